// InterOrderGCNConv_56040733278454
// MI455X (gfx1250) — compile-verified
//
#include <hip/hip_runtime.h>
#include <hip/hip_bf16.h>
#include <stdint.h>

// GCN gather-scale-scatter for gfx1250 (MI455X).
// Irregular SpMM-like workload: WMMA has no legal tiling (random edges), so
// the CDNA5 features used are the async copy engine
// (global_load_async_to_lds_b128 / ASYNCcnt / s_wait_asynccnt) for a 32-deep
// per-wave gather pipeline, and hardware global_atomic_add_f32 at device
// scope for the scatter (the L2-atomic-throughput-bound hot path).

#define FEAT   128   // feature dim (reference config)
#define BATCH  32    // edges per wave batch == async stages in flight
#define BLK    128   // 4 waves per workgroup -> 64 KB LDS/WG

// ---- guaranteed-hardware f32 atomic add (no return, device scope) ----
__device__ __forceinline__ void hw_atomic_add_f32(const float* base,
                                                  unsigned voff_bytes,
                                                  float v) {
    asm volatile("global_atomic_add_f32 %0, %1, %2 scope:SCOPE_DEV"
                 :: "v"(voff_bytes), "v"(v), "s"(base)
                 : "memory");
}

// ---------------- K0: zeroing ----------------
__global__ void zero4_kernel(float4* __restrict__ p, int n4) {
    int i = blockIdx.x * blockDim.x + threadIdx.x;
    if (i < n4) p[i] = make_float4(0.f, 0.f, 0.f, 0.f);
}

__global__ void zero1_kernel(float* __restrict__ p, int n) {
    int i = blockIdx.x * blockDim.x + threadIdx.x;
    if (i < n) p[i] = 0.f;
}

// ---------------- K1: degree accumulation ----------------
__global__ void degree_kernel(const long long* __restrict__ src,
                              const long long* __restrict__ tgt,
                              const float* __restrict__ w,
                              const float* __restrict__ deg_s,
                              const float* __restrict__ deg_t,
                              int num_edges) {
    int e = blockIdx.x * blockDim.x + threadIdx.x;
    if (e >= num_edges) return;
    float we = w[e];
    hw_atomic_add_f32(deg_s, (unsigned)src[e] * 4u, we);
    hw_atomic_add_f32(deg_t, (unsigned)tgt[e] * 4u, we);
}

// ---------------- K2: deg -> deg^-1/2 (0 if deg==0) ----------------
__global__ void invsqrt_kernel(float* __restrict__ p, int n) {
    int i = blockIdx.x * blockDim.x + threadIdx.x;
    if (i >= n) return;
    float d = p[i];
    p[i] = (d > 0.f) ? rsqrtf(d) : 0.f;
}

// ---------------- K3: batched, async-pipelined gather-scale-scatter -------
__global__ __launch_bounds__(BLK)
void gcn_scatter_kernel(const float* __restrict__ x,
                        const long long* __restrict__ src,
                        const long long* __restrict__ tgt,
                        const float* __restrict__ w,
                        const float* __restrict__ inv_s,
                        const float* __restrict__ inv_t,
                        const float* __restrict__ out,
                        int num_edges) {
    // 4 waves * 32 stages * 512 B = 64 KB LDS
    __shared__ __align__(16) float smem[(BLK / 32) * BATCH * FEAT];

    const int lane = threadIdx.x & 31;
    const int wv   = threadIdx.x >> 5;
    const int wavesPerBlk = blockDim.x >> 5;
    const int gw = blockIdx.x * wavesPerBlk + wv;
    const int nw = gridDim.x * wavesPerBlk;

    // contiguous per-wave edge chunk
    const int chunk = (num_edges + nw - 1) / nw;
    const int base  = gw * chunk;
    if (base >= num_edges) return;                 // wave-uniform exit
    const int cnt   = min(chunk, num_edges - base);
    const int limit = base + cnt;                  // this wave owns [base,limit)

    float* wbuf = &smem[wv * (BATCH * FEAT)];
    const unsigned lane16  = (unsigned)(lane * 16);
    // flat LDS-aperture pointer: low 32 bits are the LDS address
    const unsigned ldsLane = (unsigned)(uintptr_t)wbuf + lane16;

    const int nb = (cnt + BATCH - 1) / BATCH;
    for (int b = 0; b < nb; ++b) {
        // ---- per-lane edge metadata (fully coalesced) ----
        const int eRaw = base + (b << 5) + lane;
        const bool ok  = (eRaw < limit);
        const int  e   = ok ? eRaw : (num_edges - 1);   // clamped, valid
        const long long s = src[e];
        const long long t = tgt[e];
        float ew = inv_s[s] * w[e] * inv_t[t];
        if (!ok) ew = 0.f;                              // tail lanes contribute 0
        const unsigned soff = (unsigned)s << 9;         // byte offset of x row
        const unsigned toff = (unsigned)t << 9;         // byte offset of out row
        const unsigned ewBits = __float_as_uint(ew);

        // ---- issue phase: 32 async row-gathers into this wave's stages ----
        // (previous batch's ds_loads must retire before stages are reused)
        asm volatile("s_wait_dscnt 0x0" ::: "memory");
#pragma unroll
        for (int j = 0; j < BATCH; ++j) {
            const unsigned soj  = __builtin_amdgcn_readlane(soff, j);
            const unsigned voff = soj + lane16;         // lane's 16B slice
            asm volatile("global_load_async_to_lds_b128 %0, %1, %2"
                         :: "v"(ldsLane + (unsigned)(j * 512)), "v"(voff), "s"(x)
                         : "memory");
        }

        // ---- consume phase: descending ASYNCcnt ladder (in-order done) ----
#pragma unroll
        for (int j = 0; j < BATCH; ++j) {
            const unsigned toj = __builtin_amdgcn_readlane(toff, j);
            const float    ewj = __uint_as_float(__builtin_amdgcn_readlane(ewBits, j));

            asm volatile("s_wait_asynccnt %0" :: "n"(BATCH - 1 - j) : "memory");
            const float4 v =
                *reinterpret_cast<const float4*>(&wbuf[j * FEAT + lane * 4]);

            const unsigned ooff = toj + lane16;
            asm volatile("global_atomic_add_f32 %0, %1, %2 scope:SCOPE_DEV"
                         :: "v"(ooff), "v"(v.x * ewj), "s"(out) : "memory");
            asm volatile("global_atomic_add_f32 %0, %1, %2 offset:4 scope:SCOPE_DEV"
                         :: "v"(ooff), "v"(v.y * ewj), "s"(out) : "memory");
            asm volatile("global_atomic_add_f32 %0, %1, %2 offset:8 scope:SCOPE_DEV"
                         :: "v"(ooff), "v"(v.z * ewj), "s"(out) : "memory");
            asm volatile("global_atomic_add_f32 %0, %1, %2 offset:12 scope:SCOPE_DEV"
                         :: "v"(ooff), "v"(v.w * ewj), "s"(out) : "memory");
        }
    }
}

// ---------------- launch ----------------
extern "C" void kernel_launch(void* const* d_in, const int* in_sizes, int n_in,
                              void* d_out, int out_size, void* d_ws, size_t ws_size,
                              hipStream_t stream) {
    const float*     x  = (const float*)d_in[0];
    const long long* ei = (const long long*)d_in[1];   // int64 [2, E]
    const float*     w  = (const float*)d_in[2];
    // d_in[3] (num_nodes_target) stays on device; derive sizes on host:
    const int num_src   = in_sizes[0] / FEAT;
    const int num_edges = in_sizes[2];
    const int ndeg_tgt  = out_size / FEAT;

    const long long* src = ei;
    const long long* tgt = ei + num_edges;

    float* out   = (float*)d_out;
    float* deg_s = (float*)d_ws;                 // num_src floats
    float* deg_t = deg_s + num_src;              // num_tgt floats
    const int ndeg = num_src + ndeg_tgt;

    // K0: zero output (float4) and degree scratch
    {
        const int n4 = out_size / 4;
        zero4_kernel<<<(n4 + 255) / 256, 256, 0, stream>>>((float4*)out, n4);
        const int rem = out_size - n4 * 4;
        if (rem) zero1_kernel<<<1, 256, 0, stream>>>(out + n4 * 4, rem);
        zero1_kernel<<<(ndeg + 255) / 256, 256, 0, stream>>>(deg_s, ndeg);
    }

    // K1: degree accumulation (hardware f32 atomics, device scope)
    degree_kernel<<<(num_edges + 255) / 256, 256, 0, stream>>>(
        src, tgt, w, deg_s, deg_t, num_edges);

    // K2: deg -> deg^-1/2 with zero-degree guard
    invsqrt_kernel<<<(ndeg + 255) / 256, 256, 0, stream>>>(deg_s, ndeg);

    // K3: batched async gather-scale-scatter: 2048 blocks * 4 waves = 8192
    // waves, ~77 edges (~2.4 batches) each
    gcn_scatter_kernel<<<2048, BLK, 0, stream>>>(
        x, src, tgt, w, deg_s, deg_t, out, num_edges);
}